// MipCubemapEncoder_14053132993131
// MI455X (gfx1250) — compile-verified
//
#include <hip/hip_runtime.h>
#include <hip/hip_bf16.h>
#include <stdint.h>

// MipCubemapEncoder for MI455X (gfx1250).
// Memory-bound gather kernel: 432 MB streaming -> ~19us floor @ 23.3 TB/s.
// No matmul structure -> WMMA inapplicable; CDNA5 paths used instead:
//   * global_load_async_to_lds_b128 (ASYNCcnt) to stage mip levels 0+1 into LDS
//   * s_wait_asynccnt split wait counter
//   * NT temporal hints on the streaming input/output traffic so the 13 MB
//     gather tables keep L2 (192 MB) residency priority.
// Layout transform: params [6][6][L][L] -> channel-last 32B records [6][L][L][8]
// so each bilinear corner is 2x b128 instead of 6 scattered b32 gathers.

#define LVL0_L 4
#define LVL1_L 16
#define LVL2_L 64
#define LVL3_L 256

// float offsets of the transposed tables inside d_ws
#define TOFF0 0                     // 6*4*4*8     = 768 floats
#define TOFF1 768                   // 6*16*16*8   = 12288 floats
#define TOFF2 13056                 // 6*64*64*8   = 196608 floats
#define TOFF3 209664                // 6*256*256*8 = 3145728 floats
#define TTOTAL_FLOATS 3355392       // 13,421,568 bytes total
#define STAGE_FLOATS 13056          // levels 0+1 staged into LDS (52224 B)

// True clang vector type (HIP's float4 class is rejected by NT builtins).
typedef float vf4 __attribute__((ext_vector_type(4)));

// ---------------------------------------------------------------- helpers ---

struct BL { int x0, x1, y0, y1; float wx, wy; };

__device__ __forceinline__ BL bilin(float u, float v, int L) {
    float fu  = u * (float)L - 0.5f;
    float fv  = v * (float)L - 0.5f;
    float x0f = floorf(fu);
    float y0f = floorf(fv);
    int ix = (int)x0f;
    int iy = (int)y0f;
    BL b;
    b.wx = fu - x0f;
    b.wy = fv - y0f;
    b.x0 = min(max(ix,     0), L - 1);
    b.x1 = min(max(ix + 1, 0), L - 1);
    b.y0 = min(max(iy,     0), L - 1);
    b.y1 = min(max(iy + 1, 0), L - 1);
    return b;
}

__device__ __forceinline__ void face_uv(float x, float y, float z,
                                        int& face, float& u, float& v,
                                        bool& valid) {
    float ax = fabsf(x), ay = fabsf(y), az = fabsf(z);
    bool is_x = (ax >= ay) && (ax >= az);
    bool is_y = (!is_x) && (ay >= az);
    float ma = fmaxf(fmaxf(ax, ay), az);
    float sc, tc;
    if (is_x)      { face = (x >= 0.f) ? 0 : 1; sc = (x >= 0.f) ? -z : z; tc = -y; }
    else if (is_y) { face = (y >= 0.f) ? 2 : 3; sc = x; tc = (y >= 0.f) ? z : -z; }
    else           { face = (z >= 0.f) ? 4 : 5; sc = (z >= 0.f) ? x : -x; tc = -y; }
    float safe = fmaxf(ma, 1e-12f);
    u = 0.5f * (sc / safe + 1.0f);
    v = 0.5f * (tc / safe + 1.0f);
    valid = ma > 1e-12f;
}

// Sample one level from a channel-last table of 8-float (32B) records.
// Works for both LDS (inlined -> ds_load_b128) and global (global_load_b128).
// Blend uses the reference's factored lerp form for numerical fidelity.
__device__ __forceinline__ void sample_rec8(const float* __restrict__ tab,
                                            int face, float u, float v, int L,
                                            float* __restrict__ o) {
    BL b = bilin(u, v, L);
    int base = face * L * L;
    const vf4* t = (const vf4*)tab;
    int r00 = (base + b.y0 * L + b.x0) * 2;
    int r01 = (base + b.y0 * L + b.x1) * 2;
    int r10 = (base + b.y1 * L + b.x0) * 2;
    int r11 = (base + b.y1 * L + b.x1) * 2;
    vf4 A0 = t[r00], A1 = t[r00 + 1];
    vf4 B0 = t[r01], B1 = t[r01 + 1];
    vf4 C0 = t[r10], C1 = t[r10 + 1];
    vf4 D0 = t[r11], D1 = t[r11 + 1];
    float iwx = 1.f - b.wx, iwy = 1.f - b.wy;
    o[0] = (A0.x * iwx + B0.x * b.wx) * iwy + (C0.x * iwx + D0.x * b.wx) * b.wy;
    o[1] = (A0.y * iwx + B0.y * b.wx) * iwy + (C0.y * iwx + D0.y * b.wx) * b.wy;
    o[2] = (A0.z * iwx + B0.z * b.wx) * iwy + (C0.z * iwx + D0.z * b.wx) * b.wy;
    o[3] = (A0.w * iwx + B0.w * b.wx) * iwy + (C0.w * iwx + D0.w * b.wx) * b.wy;
    o[4] = (A1.x * iwx + B1.x * b.wx) * iwy + (C1.x * iwx + D1.x * b.wx) * b.wy;
    o[5] = (A1.y * iwx + B1.y * b.wx) * iwy + (C1.y * iwx + D1.y * b.wx) * b.wy;
}

// Raw LDS byte offset of a pointer into a __shared__ object (addrspace(3)).
typedef __attribute__((address_space(3))) const char lds_cchar_t;
__device__ __forceinline__ unsigned lds_off_of(const void* p) {
    return (unsigned)(unsigned long long)(lds_cchar_t*)p;
}

// NT store of 24 contiguous floats (p*96 bytes -> 16B aligned): 6x b128 NT.
__device__ __forceinline__ void store_out24_nt(float* __restrict__ out,
                                               size_t p, const float* __restrict__ o) {
    vf4* op = (vf4*)(out + p * 24);
#pragma unroll
    for (int i = 0; i < 6; ++i) {
        vf4 q = { o[4 * i + 0], o[4 * i + 1], o[4 * i + 2], o[4 * i + 3] };
        __builtin_nontemporal_store(q, op + i);
    }
}

// ---------------------------------------------------------------- kernels ---

// Transpose one level [6][6][L][L] -> [6][L][L][8] (channels padded to 8).
// L is a template arg so index math is shifts, not integer division.
// Reads are coalesced per channel (consecutive x); writes are 2x b128/thread.
template <int L>
__global__ void __launch_bounds__(256)
transpose_level(const float* __restrict__ src, float* __restrict__ dst) {
    constexpr int LL = L * L;
    int idx = blockIdx.x * 256 + threadIdx.x;
    if (idx >= 6 * LL) return;
    int f = idx / LL;                // power-of-two divide -> shift
    int r = idx - f * LL;            // y*L + x
    float vv[6];
#pragma unroll
    for (int c = 0; c < 6; ++c)
        vv[c] = src[(f * 6 + c) * LL + r];
    vf4* d = (vf4*)(dst + (size_t)idx * 8);
    vf4 lo = { vv[0], vv[1], vv[2], vv[3] };
    vf4 hi = { vv[4], vv[5], 0.f, 0.f };
    d[0] = lo;
    d[1] = hi;
}

// Main encoder, transposed-table path. Levels 0+1 async-staged into LDS.
__global__ void __launch_bounds__(256)
mip_encode_tr(const float* __restrict__ in, const float* __restrict__ ws,
              const float* __restrict__ fail, float* __restrict__ out, int n) {
    __shared__ __align__(16) float s_tab[STAGE_FLOATS];

    // CDNA5 async copy: ws[0 .. STAGE_FLOATS) -> LDS, 16B per lane per op.
    for (unsigned i = threadIdx.x * 4u; i < (unsigned)STAGE_FLOATS; i += 256u * 4u) {
        unsigned lo = lds_off_of(&s_tab[i]);
        const float* src = ws + i;
        asm volatile("global_load_async_to_lds_b128 %0, %1, off"
                     :: "v"(lo), "v"(src) : "memory");
    }
    asm volatile("s_wait_asynccnt 0" ::: "memory");
    __syncthreads();

    int p = blockIdx.x * 256 + threadIdx.x;
    if (p >= n) return;

    // Streaming input: non-temporal, read exactly once.
    float dx = __builtin_nontemporal_load(in + 3 * p + 0);
    float dy = __builtin_nontemporal_load(in + 3 * p + 1);
    float dz = __builtin_nontemporal_load(in + 3 * p + 2);

    int face; float u, v; bool valid;
    face_uv(dx, dy, dz, face, u, v, valid);

    float o[24];
    if (valid) {
        sample_rec8(&s_tab[TOFF0], face, u, v, LVL0_L, o + 0);   // LDS
        sample_rec8(&s_tab[TOFF1], face, u, v, LVL1_L, o + 6);   // LDS
        sample_rec8(ws + TOFF2,    face, u, v, LVL2_L, o + 12);  // L2
        sample_rec8(ws + TOFF3,    face, u, v, LVL3_L, o + 18);  // L2
    } else {
        float f0 = fail[0], f1 = fail[1], f2 = fail[2];
        float f3 = fail[3], f4 = fail[4], f5 = fail[5];
#pragma unroll
        for (int l = 0; l < 4; ++l) {
            o[l * 6 + 0] = f0; o[l * 6 + 1] = f1; o[l * 6 + 2] = f2;
            o[l * 6 + 3] = f3; o[l * 6 + 4] = f4; o[l * 6 + 5] = f5;
        }
    }

    store_out24_nt(out, (size_t)p, o);
}

// Fallback: original layout, scattered per-channel gathers (only if ws too small).
__global__ void __launch_bounds__(256)
mip_encode_direct(const float* __restrict__ in,
                  const float* __restrict__ q0, const float* __restrict__ q1,
                  const float* __restrict__ q2, const float* __restrict__ q3,
                  const float* __restrict__ fail, float* __restrict__ out, int n) {
    int p = blockIdx.x * 256 + threadIdx.x;
    if (p >= n) return;

    float dx = __builtin_nontemporal_load(in + 3 * p + 0);
    float dy = __builtin_nontemporal_load(in + 3 * p + 1);
    float dz = __builtin_nontemporal_load(in + 3 * p + 2);

    int face; float u, v; bool valid;
    face_uv(dx, dy, dz, face, u, v, valid);

    const float* tabs[4] = {q0, q1, q2, q3};
    const int Ls[4] = {LVL0_L, LVL1_L, LVL2_L, LVL3_L};

    float o[24];
    if (valid) {
#pragma unroll
        for (int l = 0; l < 4; ++l) {
            int L = Ls[l];
            int LL = L * L;
            BL b = bilin(u, v, L);
            const float* t = tabs[l];
            int fb = face * 6 * LL;
            float iwx = 1.f - b.wx, iwy = 1.f - b.wy;
#pragma unroll
            for (int c = 0; c < 6; ++c) {
                int cb = fb + c * LL;
                float v00 = t[cb + b.y0 * L + b.x0];
                float v01 = t[cb + b.y0 * L + b.x1];
                float v10 = t[cb + b.y1 * L + b.x0];
                float v11 = t[cb + b.y1 * L + b.x1];
                o[l * 6 + c] = (v00 * iwx + v01 * b.wx) * iwy
                             + (v10 * iwx + v11 * b.wx) * b.wy;
            }
        }
    } else {
        float f0 = fail[0], f1 = fail[1], f2 = fail[2];
        float f3 = fail[3], f4 = fail[4], f5 = fail[5];
#pragma unroll
        for (int l = 0; l < 4; ++l) {
            o[l * 6 + 0] = f0; o[l * 6 + 1] = f1; o[l * 6 + 2] = f2;
            o[l * 6 + 3] = f3; o[l * 6 + 4] = f4; o[l * 6 + 5] = f5;
        }
    }

    store_out24_nt(out, (size_t)p, o);
}

// ------------------------------------------------------------------ host ---

extern "C" void kernel_launch(void* const* d_in, const int* in_sizes, int n_in,
                              void* d_out, int out_size, void* d_ws, size_t ws_size,
                              hipStream_t stream) {
    const float* inputs = (const float*)d_in[0];
    const float* p0     = (const float*)d_in[1];
    const float* p1     = (const float*)d_in[2];
    const float* p2     = (const float*)d_in[3];
    const float* p3     = (const float*)d_in[4];
    const float* fail   = (const float*)d_in[5];
    float* out = (float*)d_out;

    int B = in_sizes[0] / 3;
    int nblk = (B + 255) / 256;

    size_t need = (size_t)TTOTAL_FLOATS * sizeof(float);
    if (d_ws != nullptr && ws_size >= need) {
        float* ws = (float*)d_ws;
        transpose_level<LVL0_L><<<(6 * LVL0_L * LVL0_L + 255) / 256, 256, 0, stream>>>(p0, ws + TOFF0);
        transpose_level<LVL1_L><<<(6 * LVL1_L * LVL1_L + 255) / 256, 256, 0, stream>>>(p1, ws + TOFF1);
        transpose_level<LVL2_L><<<(6 * LVL2_L * LVL2_L + 255) / 256, 256, 0, stream>>>(p2, ws + TOFF2);
        transpose_level<LVL3_L><<<(6 * LVL3_L * LVL3_L + 255) / 256, 256, 0, stream>>>(p3, ws + TOFF3);
        mip_encode_tr<<<nblk, 256, 0, stream>>>(inputs, ws, fail, out, B);
    } else {
        mip_encode_direct<<<nblk, 256, 0, stream>>>(inputs, p0, p1, p2, p3, fail, out, B);
    }
}